// LDPCBPDecoder_74079595921646
// MI455X (gfx1250) — compile-verified
//
#include <hip/hip_runtime.h>
#include <hip/hip_bf16.h>
#include <cstdint>

// Problem constants (fixed by the reference)
#define N_VNS_C   8192
#define N_CNS_C   4096
#define BATCH_C   1024
#define NUM_ITER_C 20
#define VDEG 3      // max variable-node degree
#define CDEG 6      // max check-node degree
#define CNB  4      // check nodes per block (async double-buffer pipeline)

__device__ __forceinline__ float phi_f(float x) {
  // phi(x) = log((e^x+1)/(e^x-1)), clipped exactly like the reference.
  // Hardware transcendentals (v_exp_f32 / v_log_f32): the kernel must stay
  // under the HBM roofline; precise OCML expf/logf would make it VALU-bound.
  x = fminf(fmaxf(x, 8.5e-08f), 16.635532f);
  float ex = __expf(x);
  return __logf(ex + 1.0f) - __logf(ex - 1.0f);
}

// ---------------- setup kernels (graph structure, run every launch) --------

__global__ void k_zero_msg(float4* __restrict__ p, size_t n4) {
  size_t i = (size_t)blockIdx.x * blockDim.x + threadIdx.x;
  size_t stride = (size_t)gridDim.x * blockDim.x;
  float4 z = {0.f, 0.f, 0.f, 0.f};
  for (; i < n4; i += stride) p[i] = z;
}

__global__ void k_zero_meta(int* __restrict__ deg, int* __restrict__ vn_edges,
                            int* __restrict__ cn_start, int* __restrict__ cn_end) {
  int i = blockIdx.x * blockDim.x + threadIdx.x;
  if (i < N_VNS_C) deg[i] = 0;
  if (i < N_VNS_C * VDEG) vn_edges[i] = -1;
  if (i < N_CNS_C) { cn_start[i] = 0; cn_end[i] = 0; }
}

__global__ void k_fill(const int* __restrict__ vn_con, const int* __restrict__ cn_con,
                       int E, int* __restrict__ deg, int* __restrict__ vn_edges,
                       int* __restrict__ cn_start, int* __restrict__ cn_end) {
  int e = blockIdx.x * blockDim.x + threadIdx.x;
  if (e >= E) return;
  int v = vn_con[e];
  int slot = atomicAdd(&deg[v], 1);
  if (slot < VDEG) vn_edges[v * VDEG + slot] = e;
  // edges are sorted by cn -> branch-free contiguous-range detection
  int c = cn_con[e];
  if (e == 0 || cn_con[e - 1] != c) cn_start[c] = e;
  if (e == E - 1 || cn_con[e + 1] != c) cn_end[c] = e + 1;
}

// canonicalize per-VN edge lists (atomicAdd order is nondeterministic) by
// sorting the <=3 entries ascending; -1 (0xFFFFFFFF unsigned) sorts last.
__global__ void k_sortfix(int* __restrict__ vn_edges) {
  int v = blockIdx.x * blockDim.x + threadIdx.x;
  if (v >= N_VNS_C) return;
  int x0 = vn_edges[v * VDEG + 0];
  int x1 = vn_edges[v * VDEG + 1];
  int x2 = vn_edges[v * VDEG + 2];
  int t;
  if ((unsigned)x0 > (unsigned)x1) { t = x0; x0 = x1; x1 = t; }
  if ((unsigned)x1 > (unsigned)x2) { t = x1; x1 = x2; x2 = t; }
  if ((unsigned)x0 > (unsigned)x1) { t = x0; x0 = x1; x1 = t; }
  vn_edges[v * VDEG + 0] = x0;
  vn_edges[v * VDEG + 1] = x1;
  vn_edges[v * VDEG + 2] = x2;
}

// out[c][r] = -in[r][c], in is R x C (both multiples of 32). Tiled via LDS.
__global__ void k_transpose_neg(const float* __restrict__ in, float* __restrict__ out,
                                int R, int C) {
  __shared__ float tile[32][33];
  int c0 = blockIdx.x * 32, r0 = blockIdx.y * 32;
  int x = threadIdx.x;
  for (int y = threadIdx.y; y < 32; y += blockDim.y)
    tile[y][x] = in[(size_t)(r0 + y) * C + c0 + x];
  __syncthreads();
  for (int y = threadIdx.y; y < 32; y += blockDim.y)
    out[(size_t)(c0 + y) * R + r0 + x] = -tile[x][y];
}

// ---------------- per-iteration kernels ------------------------------------

// S[v][:] = llrT[v][:] + sum_{e in N(v)} msg[e][:]
// One block per VN; 256 threads x float4 = full 1024-wide batch row.
__global__ void __launch_bounds__(256)
k_vnsum(const float* __restrict__ msg, const float* __restrict__ llrT,
        const int* __restrict__ vn_edges, float* __restrict__ S) {
  const int v = blockIdx.x;
  const int col = threadIdx.x * 4;
  float4 acc = *(const float4*)(llrT + ((size_t)v << 10) + col);
#pragma unroll
  for (int k = 0; k < VDEG; ++k) {
    int e = vn_edges[v * VDEG + k];          // uniform -> scalar load
    if (e >= 0) {
      float4 m = *(const float4*)(msg + ((size_t)e << 10) + col);
      acc.x += m.x; acc.y += m.y; acc.z += m.z; acc.w += m.w;
    }
  }
  *(float4*)(S + ((size_t)v << 10) + col) = acc;
}

// CN update (boxplus-phi), in-place on msg. Each block handles CNB consecutive
// check nodes, double-buffering their contiguous message rows through LDS with
// gfx1250 async global->LDS copies (ASYNCcnt pipeline). Each thread consumes
// exactly the chunk it staged, so s_wait_asynccnt (wave-scope) suffices.
__global__ void __launch_bounds__(256)
k_cn_update(float* __restrict__ msg, const float* __restrict__ S,
            const int* __restrict__ vn_con, const int* __restrict__ cn_start,
            const int* __restrict__ cn_end) {
  __shared__ float smem[2 * CDEG * BATCH_C];  // 48 KB
  const int t = threadIdx.x;
  const int col = t * 4;
  const int cbase = blockIdx.x * CNB;

  auto prefetch = [&](int c, int buf) {
    int e0 = cn_start[c];
    int dg = cn_end[c] - e0;
#pragma unroll
    for (int k = 0; k < CDEG; ++k) {
      int e = e0 + ((k < dg) ? k : 0);  // pad short CNs: reload row 0 (fixed count)
      uint32_t lds = (uint32_t)(uintptr_t)&smem[(buf * CDEG + k) * BATCH_C + col];
      uint64_t g = (uint64_t)(uintptr_t)(msg + ((size_t)e << 10) + col);
      asm volatile("global_load_async_to_lds_b128 %0, %1, off"
                   :: "v"(lds), "v"(g) : "memory");
    }
  };

  prefetch(cbase, 0);
#pragma unroll 1
  for (int i = 0; i < CNB; ++i) {
    if (i + 1 < CNB) {
      prefetch(cbase + i + 1, (i + 1) & 1);          // 6 more in flight
      asm volatile("s_wait_asynccnt 6" ::: "memory"); // current group landed
    } else {
      asm volatile("s_wait_asynccnt 0" ::: "memory");
    }
    const int c = cbase + i;
    const int e0 = cn_start[c];
    const int dg = cn_end[c] - e0;
    const float* lbuf = &smem[(i & 1) * CDEG * BATCH_C + col];

    float p[CDEG][4];
    float psum[4] = {0.f, 0.f, 0.f, 0.f};
    uint32_t sgn = 0;  // bit (k*4+j): x_{k,j} < 0
#pragma unroll
    for (int k = 0; k < CDEG; ++k) {
      if (k < dg) {
        int e = e0 + k;
        int v = vn_con[e];                                    // uniform scalar
        float4 sv = *(const float4*)(S + ((size_t)v << 10) + col);
        float4 mv = *(const float4*)(lbuf + k * BATCH_C);
        float xs[4] = {sv.x - mv.x, sv.y - mv.y, sv.z - mv.z, sv.w - mv.w};
#pragma unroll
        for (int j = 0; j < 4; ++j) {
          sgn |= (uint32_t)(xs[j] < 0.0f) << (k * 4 + j);
          float pv = phi_f(fabsf(xs[j]));
          p[k][j] = pv;
          psum[j] += pv;
        }
      }
    }
    uint32_t par = 0;  // per-component sign-product parity (bits 0..3)
#pragma unroll
    for (int j = 0; j < 4; ++j)
      par |= (uint32_t)(__popc(sgn & (0x11111111u << j)) & 1) << j;

#pragma unroll
    for (int k = 0; k < CDEG; ++k) {
      if (k < dg) {
        int e = e0 + k;
        float ov[4];
#pragma unroll
        for (int j = 0; j < 4; ++j) {
          float r = phi_f(psum[j] - p[k][j]);
          uint32_t neg = ((sgn >> (k * 4 + j)) & 1u) ^ ((par >> j) & 1u);
          ov[j] = neg ? -r : r;
        }
        float4 o = {ov[0], ov[1], ov[2], ov[3]};
        *(float4*)(msg + ((size_t)e << 10) + col) = o;
      }
    }
  }
}

// ---------------- driver ----------------------------------------------------

extern "C" void kernel_launch(void* const* d_in, const int* in_sizes, int n_in,
                              void* d_out, int out_size, void* d_ws, size_t ws_size,
                              hipStream_t stream) {
  (void)n_in; (void)out_size; (void)ws_size;
  const float* llr_ch = (const float*)d_in[0];
  const int* vn_con = (const int*)d_in[1];
  const int* cn_con = (const int*)d_in[2];
  const int E = in_sizes[1];  // num_edges (<= N_VNS*VDEG)

  auto al = [](size_t x) { return (x + 255) & ~(size_t)255; };
  char* ws = (char*)d_ws;
  size_t off = 0;
  float* msg  = (float*)(ws + off); off += al((size_t)E * BATCH_C * sizeof(float));
  float* S    = (float*)(ws + off); off += al((size_t)N_VNS_C * BATCH_C * sizeof(float));
  float* llrT = (float*)(ws + off); off += al((size_t)N_VNS_C * BATCH_C * sizeof(float));
  int* deg      = (int*)(ws + off); off += al(N_VNS_C * sizeof(int));
  int* vn_edges = (int*)(ws + off); off += al((size_t)N_VNS_C * VDEG * sizeof(int));
  int* cn_s     = (int*)(ws + off); off += al(N_CNS_C * sizeof(int));
  int* cn_e     = (int*)(ws + off); off += al(N_CNS_C * sizeof(int));

  // build graph structure + zero message state (every call; deterministic)
  size_t n4 = (size_t)E * BATCH_C / 4;
  k_zero_msg<<<4096, 256, 0, stream>>>((float4*)msg, n4);
  k_zero_meta<<<(N_VNS_C * VDEG + 255) / 256, 256, 0, stream>>>(deg, vn_edges, cn_s, cn_e);
  k_fill<<<(E + 255) / 256, 256, 0, stream>>>(vn_con, cn_con, E, deg, vn_edges, cn_s, cn_e);
  k_sortfix<<<(N_VNS_C + 255) / 256, 256, 0, stream>>>(vn_edges);

  {  // llrT[v][b] = -llr_ch[b][v]
    dim3 g(N_VNS_C / 32, BATCH_C / 32), b(32, 8);
    k_transpose_neg<<<g, b, 0, stream>>>(llr_ch, llrT, BATCH_C, N_VNS_C);
  }

  for (int it = 0; it < NUM_ITER_C; ++it) {
    k_vnsum<<<N_VNS_C, 256, 0, stream>>>(msg, llrT, vn_edges, S);
    k_cn_update<<<N_CNS_C / CNB, 256, 0, stream>>>(msg, S, vn_con, cn_s, cn_e);
  }

  // x_tot = llr + segment_sum(msg); out = -x_tot^T
  k_vnsum<<<N_VNS_C, 256, 0, stream>>>(msg, llrT, vn_edges, S);
  {
    dim3 g(BATCH_C / 32, N_VNS_C / 32), b(32, 8);
    k_transpose_neg<<<g, b, 0, stream>>>(S, (float*)d_out, N_VNS_C, BATCH_C);
  }
}